// TSMNet_24129126269567
// MI455X (gfx1250) — compile-verified
//
#include <hip/hip_runtime.h>
#include <hip/hip_bf16.h>

// ---------------- problem constants ----------------
#define TDIM   1000
#define BATCH  128
#define CCH    32
#define F1N    8
#define F2N    40
#define SUBN   20
#define KSZ    25
#define PADK   12          // (K-1)/2
#define CKLEN  (CCH*KSZ)   // 800
#define NCLS   4
#define NTRIU  210
#define REEIG_EPS 1e-4f
#define BN_EPS    1e-5f

// padded layouts (remove all load/store guards in WMMA kernels)
#define H2R 48             // h2 rows per batch (40 used)
#define H2C 1008           // h2 cols (1000 used)
#define YR  32             // y rows per batch (20 used, 20..31 exact zeros)
#define YC  1008           // y cols (1000 used)

typedef __attribute__((ext_vector_type(2))) float v2f;
typedef __attribute__((ext_vector_type(8))) float v8f;

// fp32 WMMA 16x16x4: D = A(16x4) * B(4x16) + C, full fp32 precision path.
#define WMMA_F32X4(a, b, c) \
  __builtin_amdgcn_wmma_f32_16x16x4_f32(false, (a), false, (b), (short)0, (c), false, false)

// ---------------- fused conv weights: weff[f2][c*25+k] = sum_f1 w2[f2,f1,c]*w1[f1,k] ----------------
__global__ void k_weff(const float* __restrict__ w1, const float* __restrict__ b1,
                       const float* __restrict__ w2, const float* __restrict__ b2,
                       float* __restrict__ weff, float* __restrict__ beff) {
  int f2 = blockIdx.x;
  for (int ck = threadIdx.x; ck < CKLEN; ck += blockDim.x) {
    int c = ck / KSZ, k = ck - c * KSZ;
    float s = 0.f;
    for (int f1 = 0; f1 < F1N; ++f1)
      s += w2[(f2 * F1N + f1) * CCH + c] * w1[f1 * KSZ + k];
    weff[f2 * CKLEN + ck] = s;
  }
  if (threadIdx.x == 0) {
    float s = b2[f2];
    for (int f1 = 0; f1 < F1N; ++f1) {
      float cs = 0.f;
      for (int c = 0; c < CCH; ++c) cs += w2[(f2 * F1N + f1) * CCH + c];
      s += cs * b1[f1];
    }
    beff[f2] = s;
  }
}

// ---------------- conv as GEMM via WMMA: h2[b][f2][t] = sum_ck weff[f2][ck] * x_reflect ----------------
// grid (63, B), block 96 = 3 waves (wave w handles f2-tile w). No guards: h2 padded 48x1008.
__global__ __launch_bounds__(96) void k_conv(const float* __restrict__ x,
                                             const float* __restrict__ weff,
                                             const float* __restrict__ beff,
                                             float* __restrict__ h2) {
  __shared__ float xw[CCH * 40];            // 32 channels x (16 + 24) taps, reflect-padded
  int ttile = blockIdx.x;
  int b = blockIdx.y;
  int tid = threadIdx.x;
  int tbase = ttile * 16 - PADK;
  for (int idx = tid; idx < CCH * 40; idx += 96) {
    int c = idx / 40, u = idx - c * 40;
    int t = tbase + u;
    if (t < 0) t = -t;
    if (t >= TDIM) t = 2 * TDIM - 2 - t;    // reflect
    xw[idx] = x[(b * CCH + c) * TDIM + t];
  }
  __syncthreads();

  int lane = tid & 31;
  int mtile = tid >> 5;                      // 0..2
  int half = lane >> 4;
  int l16 = lane & 15;
  int m = mtile * 16 + l16;                  // 0..47; rows >=40 compute garbage, stored to pad rows
  const float* wrow = weff + (m < F2N ? m : F2N - 1) * CKLEN;  // clamp: valid address always

  // carry-tracked decomposition of kk0 = kb + half*2 into (c0, r0): r0 = kk0 % 25
  int r0 = half * 2;                         // kk0 < 25 initially -> c0 = 0
  int idx0 = r0 + l16;                       // c0*40 + r0 + l16
  v8f acc = {};
#pragma unroll 4
  for (int kb = 0; kb < CKLEN; kb += 4) {
    int idx1 = (r0 == 24) ? (idx0 + 16) : (idx0 + 1);   // kk0+1: wrap -24, +40 channel
    v2f a = *(const v2f*)(wrow + kb + half * 2);        // global_load_b64
    v2f bb;
    bb.x = xw[idx0];
    bb.y = xw[idx1];
    acc = WMMA_F32X4(a, bb, acc);
    r0 += 4; idx0 += 4;
    if (r0 >= 25) { r0 -= 25; idx0 += 15; }             // -25 + 40
  }
  int col = ttile * 16 + l16;                // <= 1007, inside padded cols
  float bof = beff[m < F2N ? m : 0];
#pragma unroll
  for (int r = 0; r < 8; ++r) {
    int row = mtile * 16 + r + half * 8;     // 0..47
    h2[((size_t)b * H2R + row) * H2C + col] = acc[r] + bof;
  }
}

// ---------------- y = W^T * h2  (M=20->32 padded, K=40, N=1000->1008) via WMMA ----------------
// grid (63, B), block 64 = 2 waves (wave = mtile). Rows 20..31 of y written as exact zeros.
__global__ __launch_bounds__(64) void k_proj(const float* __restrict__ h2,
                                             const float* __restrict__ W,
                                             float* __restrict__ y) {
  int ttile = blockIdx.x, b = blockIdx.y;
  int tid = threadIdx.x, lane = tid & 31, mtile = tid >> 5;
  int half = lane >> 4, l16 = lane & 15;
  int m = mtile * 16 + l16;                  // 0..31
  bool mok = m < SUBN;
  int mc = mok ? m : SUBN - 1;               // clamped address
  int t = ttile * 16 + l16;                  // <= 1007, valid in padded h2/y
  const float* hb = h2 + (size_t)b * H2R * H2C;
  v8f acc = {};
#pragma unroll
  for (int kb = 0; kb < F2N; kb += 4) {
    int kk0 = kb + half * 2;
    v2f a, bb;
    a.x = mok ? W[kk0 * SUBN + mc] : 0.f;          // cndmask select, unconditional load
    a.y = mok ? W[(kk0 + 1) * SUBN + mc] : 0.f;
    bb.x = hb[kk0 * H2C + t];
    bb.y = hb[(kk0 + 1) * H2C + t];
    acc = WMMA_F32X4(a, bb, acc);
  }
#pragma unroll
  for (int r = 0; r < 8; ++r) {
    int row = mtile * 16 + r + half * 8;     // 0..31
    y[((size_t)b * YR + row) * YC + t] = acc[r];
  }
}

// ---------------- center y rows over T (one wave per (b,m) row, m<20) ----------------
__global__ __launch_bounds__(32) void k_center(float* __restrict__ y) {
  int b = blockIdx.x / SUBN, m = blockIdx.x - b * SUBN;
  float* p = y + ((size_t)b * YR + m) * YC;
  int lane = threadIdx.x;
  float s = 0.f;
  for (int t = lane; t < TDIM; t += 32) s += p[t];
  for (int off = 16; off; off >>= 1) s += __shfl_down(s, off, 32);
  s = __shfl(s, 0, 32);
  float mean = s * (1.0f / TDIM);
  for (int t = lane; t < TDIM; t += 32) p[t] -= mean;
}

// ---------------- S = yc*yc^T/(T-1)  (M=N=20, K=1000) via WMMA; y padded -> no load guards ----------------
// grid B, block 128 = 4 waves (wave = (mtile,ntile))
__global__ __launch_bounds__(128) void k_gram(const float* __restrict__ y,
                                              float* __restrict__ S) {
  int b = blockIdx.x;
  int tid = threadIdx.x, lane = tid & 31, w = tid >> 5;
  int mtile = w >> 1, ntile = w & 1;
  int half = lane >> 4, l16 = lane & 15;
  int m = mtile * 16 + l16;                  // 0..31 all valid (rows >=20 are zeros)
  int n = ntile * 16 + l16;
  const float* yb = y + (size_t)b * YR * YC;
  const float* am = yb + m * YC;
  const float* bn = yb + n * YC;
  v8f acc = {};
#pragma unroll 5
  for (int kb = 0; kb < TDIM; kb += 4) {
    int kk0 = kb + half * 2;
    v2f a  = *(const v2f*)(am + kk0);        // global_load_b64
    v2f bb = *(const v2f*)(bn + kk0);        // B[k][n] = y[n][k]
    acc = WMMA_F32X4(a, bb, acc);
  }
  const float inv = 1.0f / (TDIM - 1);
#pragma unroll
  for (int r = 0; r < 8; ++r) {
    int row = mtile * 16 + r + half * 8;
    int col = ntile * 16 + l16;
    if (row < SUBN && col < SUBN)
      S[b * SUBN * SUBN + row * SUBN + col] = acc[r] * inv;
  }
}

// ---------------- batched 20x20 symmetric eig (cyclic Jacobi) + spectral function ----------------
#define EIG_CLAMP    0   // out1 = V max(w,eps) V^T
#define EIG_LOG      1   // out1 = V log(w) V^T   (+ optional sumsq = sum log(w)^2 = ||logm||_F^2)
#define EIG_EXP      2   // out1 = V exp(w) V^T
#define EIG_SQRTPAIR 3   // out1 = V sqrt(w) V^T, out2 = V (1/sqrt(w)) V^T

__global__ __launch_bounds__(32) void k_eig(const float* __restrict__ Xin,
                                            const float* __restrict__ P,   // optional: A = P*X*P
                                            float* __restrict__ out1,
                                            float* __restrict__ out2,
                                            float* __restrict__ sumsq,
                                            int mode) {
  __shared__ float A[SUBN][SUBN + 1];
  __shared__ float V[SUBN][SUBN + 1];
  __shared__ float T1[SUBN][SUBN + 1];
  __shared__ float Pm[SUBN][SUBN + 1];
  __shared__ float fw[SUBN], gw[SUBN];
  int bmat = blockIdx.x;
  int j = threadIdx.x;
  const float* X = Xin + bmat * SUBN * SUBN;

  if (j < SUBN) {
    for (int i = 0; i < SUBN; ++i) A[i][j] = X[i * SUBN + j];
    if (P) for (int i = 0; i < SUBN; ++i) Pm[i][j] = P[i * SUBN + j];
  }
  __syncthreads();

  if (P) {  // congruence sandwich A = P * X * P
    if (j < SUBN)
      for (int i = 0; i < SUBN; ++i) {
        float s = 0.f;
        for (int k = 0; k < SUBN; ++k) s += Pm[i][k] * A[k][j];
        T1[i][j] = s;
      }
    __syncthreads();
    if (j < SUBN)
      for (int i = 0; i < SUBN; ++i) {
        float s = 0.f;
        for (int k = 0; k < SUBN; ++k) s += T1[i][k] * Pm[k][j];
        A[i][j] = s;
      }
    __syncthreads();
  }

  // symmetrize (via temp to avoid cross-lane race) and init V = I
  if (j < SUBN)
    for (int i = 0; i < SUBN; ++i) T1[i][j] = 0.5f * (A[i][j] + A[j][i]);
  __syncthreads();
  if (j < SUBN)
    for (int i = 0; i < SUBN; ++i) { A[i][j] = T1[i][j]; V[i][j] = (i == j) ? 1.f : 0.f; }
  __syncthreads();

  // cyclic Jacobi: 9 sweeps, rotation applied lane-parallel over 20 columns
  for (int sweep = 0; sweep < 9; ++sweep) {
    for (int p = 0; p < SUBN - 1; ++p) {
      for (int q = p + 1; q < SUBN; ++q) {
        float apq = A[p][q];                 // uniform across wave (same LDS address)
        float app = A[p][p], aqq = A[q][q];
        if (fabsf(apq) > 1e-13f) {
          float theta = 0.5f * (aqq - app) / apq;
          float tt = ((theta >= 0.f) ? 1.f : -1.f) / (fabsf(theta) + sqrtf(theta * theta + 1.f));
          float c = rsqrtf(tt * tt + 1.f);
          float s = tt * c;
          if (j < SUBN) {                    // rows p,q : J^T A
            float apj = A[p][j], aqj = A[q][j];
            A[p][j] = c * apj - s * aqj;
            A[q][j] = s * apj + c * aqj;
          }
          __syncthreads();
          if (j < SUBN) {                    // cols p,q : (.)J ; V = V J
            float ajp = A[j][p], ajq = A[j][q];
            A[j][p] = c * ajp - s * ajq;
            A[j][q] = s * ajp + c * ajq;
            float vjp = V[j][p], vjq = V[j][q];
            V[j][p] = c * vjp - s * vjq;
            V[j][q] = s * vjp + c * vjq;
          }
          __syncthreads();
        }
      }
    }
  }

  if (j < SUBN) {
    float wv = A[j][j];
    float f, g = 0.f;
    if (mode == EIG_CLAMP)      f = fmaxf(wv, REEIG_EPS);
    else if (mode == EIG_LOG)   f = logf(fmaxf(wv, 1e-30f));
    else if (mode == EIG_EXP)   f = expf(wv);
    else { float wp = fmaxf(wv, 1e-30f); f = sqrtf(wp); g = 1.f / f; }
    fw[j] = f; gw[j] = g;
  }
  __syncthreads();

  if (sumsq && j == 0) {     // ||f(X)||_F^2 = sum f(w)^2 (Frobenius invariance)
    float s = 0.f;
    for (int k = 0; k < SUBN; ++k) s += fw[k] * fw[k];
    sumsq[bmat] = s;
  }

  if (j < SUBN) {
    for (int i = 0; i < SUBN; ++i) {
      float s1 = 0.f, s2 = 0.f;
      for (int k = 0; k < SUBN; ++k) {
        float vik = V[i][k], vjk = V[j][k];
        s1 += vik * fw[k] * vjk;
        if (mode == EIG_SQRTPAIR) s2 += vik * gw[k] * vjk;
      }
      out1[bmat * SUBN * SUBN + i * SUBN + j] = s1;
      if (out2) out2[bmat * SUBN * SUBN + i * SUBN + j] = s2;
    }
  }
}

// ---------------- out = P * X * P (single 20x20) ----------------
__global__ __launch_bounds__(32) void k_sandwich(const float* __restrict__ Pp,
                                                 const float* __restrict__ X,
                                                 float* __restrict__ out) {
  __shared__ float Pm[SUBN][SUBN + 1], Xm[SUBN][SUBN + 1], T1[SUBN][SUBN + 1];
  int j = threadIdx.x;
  if (j < SUBN)
    for (int i = 0; i < SUBN; ++i) { Pm[i][j] = Pp[i * SUBN + j]; Xm[i][j] = X[i * SUBN + j]; }
  __syncthreads();
  if (j < SUBN)
    for (int i = 0; i < SUBN; ++i) {
      float s = 0.f;
      for (int k = 0; k < SUBN; ++k) s += Pm[i][k] * Xm[k][j];
      T1[i][j] = s;
    }
  __syncthreads();
  if (j < SUBN)
    for (int i = 0; i < SUBN; ++i) {
      float s = 0.f;
      for (int k = 0; k < SUBN; ++k) s += T1[i][k] * Pm[k][j];
      out[i * SUBN + j] = s;
    }
}

// ---------------- deterministic mean over batch ----------------
__global__ void k_reduce_mean(const float* __restrict__ in, float* __restrict__ out,
                              int nb, int len) {
  for (int e = blockIdx.x * blockDim.x + threadIdx.x; e < len; e += gridDim.x * blockDim.x) {
    float s = 0.f;
    for (int b = 0; b < nb; ++b) s += in[b * len + e];
    out[e] = s / (float)nb;
  }
}

// ---------------- var -> power scalar ----------------
__global__ __launch_bounds__(128) void k_power(const float* __restrict__ sumsq,
                                               const float* __restrict__ bn_std,
                                               float* __restrict__ power) {
  __shared__ float red[128];
  int t = threadIdx.x;
  red[t] = sumsq[t];
  __syncthreads();
  for (int off = 64; off; off >>= 1) { if (t < off) red[t] += red[t + off]; __syncthreads(); }
  if (t == 0) {
    float var = red[0] / (float)BATCH;
    power[0] = bn_std[0] * rsqrtf(var + BN_EPS);
  }
}

// ---------------- final: logits = (power * triu(Lc) * scale) @ lin_w^T + lin_b ----------------
__global__ __launch_bounds__(32) void k_logits(const float* __restrict__ Lc,
                                               const float* __restrict__ power,
                                               const float* __restrict__ lin_w,
                                               const float* __restrict__ lin_b,
                                               float* __restrict__ out) {
  int b = blockIdx.x, lane = threadIdx.x;
  float pw = power[0];
  float a0 = 0.f, a1 = 0.f, a2 = 0.f, a3 = 0.f;
  for (int e = lane; e < NTRIU; e += 32) {
    int i = 0, off = 0;
    while (off + (SUBN - i) <= e) { off += SUBN - i; ++i; }
    int jj = i + (e - off);
    float scale = (i == jj) ? 1.0f : 1.41421356237309515f;
    float val = pw * Lc[b * SUBN * SUBN + i * SUBN + jj] * scale;
    a0 += val * lin_w[0 * NTRIU + e];
    a1 += val * lin_w[1 * NTRIU + e];
    a2 += val * lin_w[2 * NTRIU + e];
    a3 += val * lin_w[3 * NTRIU + e];
  }
  for (int off = 16; off; off >>= 1) {
    a0 += __shfl_down(a0, off, 32);
    a1 += __shfl_down(a1, off, 32);
    a2 += __shfl_down(a2, off, 32);
    a3 += __shfl_down(a3, off, 32);
  }
  if (lane == 0) {
    out[b * NCLS + 0] = a0 + lin_b[0];
    out[b * NCLS + 1] = a1 + lin_b[1];
    out[b * NCLS + 2] = a2 + lin_b[2];
    out[b * NCLS + 3] = a3 + lin_b[3];
  }
}

// ---------------- host orchestration (graph-capture safe: all on stream) ----------------
extern "C" void kernel_launch(void* const* d_in, const int* in_sizes, int n_in,
                              void* d_out, int out_size, void* d_ws, size_t ws_size,
                              hipStream_t stream) {
  (void)in_sizes; (void)n_in; (void)out_size; (void)ws_size;
  const float* x     = (const float*)d_in[0];
  // d_in[1] = domains (unused by reference)
  const float* w1    = (const float*)d_in[2];
  const float* b1    = (const float*)d_in[3];
  const float* w2    = (const float*)d_in[4];
  const float* b2    = (const float*)d_in[5];
  const float* Wb    = (const float*)d_in[6];
  const float* bnstd = (const float*)d_in[7];
  const float* lw    = (const float*)d_in[8];
  const float* lb    = (const float*)d_in[9];

  float* ws   = (float*)d_ws;
  float* weff = ws;                        // 32000
  float* beff = weff + 32000;              // 40
  float* h2   = ws + 32768;                // B*48*1008 = 6,193,152
  float* yb   = h2 + (size_t)BATCH*H2R*H2C;   // B*32*1008 = 4,128,768 (float-offset even -> 8B aligned)
  float* S    = yb + (size_t)BATCH*YR*YC;     // B*400
  float* Lt   = S  + BATCH * 400;          // B*400
  float* G    = Lt + BATCH * 400;          // 400
  float* Gs   = G   + 400;
  float* Gis  = Gs  + 400;
  float* Mt   = Gis + 400;
  float* Em   = Mt  + 400;
  float* ssq  = Em  + 400;                 // 128
  float* powr = ssq + BATCH;               // 1

  k_weff<<<F2N, 128, 0, stream>>>(w1, b1, w2, b2, weff, beff);
  k_conv<<<dim3(63, BATCH), 96, 0, stream>>>(x, weff, beff, h2);
  k_proj<<<dim3(63, BATCH), 64, 0, stream>>>(h2, Wb, yb);
  k_center<<<BATCH * SUBN, 32, 0, stream>>>(yb);
  k_gram<<<BATCH, 128, 0, stream>>>(yb, S);

  // ReEig
  k_eig<<<BATCH, 32, 0, stream>>>(S, nullptr, S, nullptr, nullptr, EIG_CLAMP);
  // G = mean(S)
  k_reduce_mean<<<1, 512, 0, stream>>>(S, G, BATCH, SUBN * SUBN);

  // Karcher mean iterations
  for (int it = 0; it < 3; ++it) {
    k_eig<<<1, 32, 0, stream>>>(G, nullptr, Gs, Gis, nullptr, EIG_SQRTPAIR);
    k_eig<<<BATCH, 32, 0, stream>>>(S, Gis, Lt, nullptr, nullptr, EIG_LOG);
    k_reduce_mean<<<1, 512, 0, stream>>>(Lt, Mt, BATCH, SUBN * SUBN);
    k_eig<<<1, 32, 0, stream>>>(Mt, nullptr, Em, nullptr, nullptr, EIG_EXP);
    k_sandwich<<<1, 32, 0, stream>>>(Gs, Em, G);
  }

  // final whitening + logm (Lc), variance, power; logm(Sn) = power * Lc
  k_eig<<<1, 32, 0, stream>>>(G, nullptr, Gs, Gis, nullptr, EIG_SQRTPAIR);
  k_eig<<<BATCH, 32, 0, stream>>>(S, Gis, Lt, nullptr, ssq, EIG_LOG);
  k_power<<<1, 128, 0, stream>>>(ssq, bnstd, powr);
  k_logits<<<BATCH, 32, 0, stream>>>(Lt, powr, lw, lb, (float*)d_out);
}